// GraphConvolution_17076789969202
// MI455X (gfx1250) — compile-verified
//
#include <hip/hip_runtime.h>

// ---------------------------------------------------------------------------
// R-GCN layer for MI455X (gfx1250, wave32, WMMA).
//
// out[dst] += x[src] @ W_r  ==  out[dst] += (x @ W_r)[src]
// => 9 dense GEMMs (50000x256x256, fp32 WMMA) + 8 edge scatter-adds.
// Compute: 216 GFLOP -> 59 GFLOP. Scatter (~3.3 GB) resolves in the 192 MB L2
// (y_r = 51 MB, out = 51 MB both resident). HBM: ~0.5 GB total -> ~25 us floor.
//
// GEMM: each block owns a 128-column half of W, staged transposed in LDS
// (133 KB, stride 260 -> conflict-free b64 reads) => 2 blocks/WGP, 4 waves/
// SIMD for latency hiding. Wave tile 32Mx128N: each B fragment feeds 2 WMMAs
// (16 FLOP per LDS byte), A loads batched 4 k-steps ahead, B fragments
// batched 8-wide so ds latency is amortized across whole WMMA groups.
// ---------------------------------------------------------------------------

typedef float v2f __attribute__((ext_vector_type(2)));
typedef float v8f __attribute__((ext_vector_type(8)));

#define DDIM       256     // D_IN == D_OUT
#define NHALF      128     // N columns per block (half of W)
#define LDS_STRIDE 260     // floats per LDS row; 260 % 64 == 4 -> conflict-free
#define STRH       130     // v2f elements per LDS row
#define LDS_BYTES  (NHALF * LDS_STRIDE * 4)   // 133120 B -> 2 blocks per WGP

static __device__ __forceinline__ v8f wmma_f32(v2f a, v2f b, v8f c) {
    return __builtin_amdgcn_wmma_f32_16x16x4_f32(
        /*neg_a=*/false, a, /*neg_b=*/false, b,
        /*c_mod=*/(short)0, c, /*reuse_a=*/false, /*reuse_b=*/false);
}

// Y[m,n] = sum_k X[m,k] * W[k,n].  M rows, K = N = 256.
// Grid: (ceil(M/256), 2). Block: 256 threads = 8 waves.
// Block tile: 256M x 128N (N half = blockIdx.y). Wave w: rows +32w, all 128 N.
extern "C" __global__ __launch_bounds__(256)
void rgcn_gemm_wmma(const float* __restrict__ X,
                    const float* __restrict__ W,
                    float* __restrict__ Y,
                    int M)
{
    extern __shared__ char smem[];
    float*     ldsW  = (float*)smem;        // [n_local][k], row stride LDS_STRIDE
    const v2f* ldsW2 = (const v2f*)smem;    // same memory, v2f view (b64 loads)

    const int tid   = threadIdx.x;
    const int nCol0 = blockIdx.y * NHALF;   // global N base of this block

    // Stage this block's half of W (row-major [k][n]) transposed into LDS as
    // [n_local][k]. Thread t: n_local = t&127, k half = (t>>7)*128.
    {
        const int nl   = tid & (NHALF - 1);
        const int kOff = (tid >> 7) * (DDIM / 2);
        for (int k = 0; k < DDIM / 2; ++k) {
            ldsW[nl * LDS_STRIDE + kOff + k] = W[(kOff + k) * DDIM + nCol0 + nl];
        }
    }
    __syncthreads();

    const int wave  = tid >> 5;
    const int lane  = tid & 31;
    const int ln    = lane & 15;            // fragment row (A) / col (B,C)
    const int h     = lane >> 4;            // lane-half -> K sub-offset 2h
    const int mBase = blockIdx.x * 256 + wave * 32;

    // 16 accumulators = 32M x 128N strip. acc[s*8+nb]: rows +16s, cols +16nb.
    v8f acc[16];
#pragma unroll
    for (int i = 0; i < 16; ++i) acc[i] = (v8f)0.0f;

    const int  m0  = mBase + ln,      m1  = mBase + 16 + ln;
    const bool ok0 = (m0 < M),        ok1 = (m1 < M);
    const int  mc0 = ok0 ? m0 : M - 1;      // clamped: load always legal,
    const int  mc1 = ok1 ? m1 : M - 1;      // zeroed via cndmask (no exec churn)
    const float* __restrict__ X0 = X + (size_t)mc0 * DDIM;
    const float* __restrict__ X1 = X + (size_t)mc1 * DDIM;

    for (int kk = 0; kk < DDIM; kk += 16) {     // 16-wide K chunk = 4 WMMA steps
        // Batch A loads for 4 k-steps (8 independent global_load_b64).
        v2f a0[4], a1[4];
#pragma unroll
        for (int j = 0; j < 4; ++j) {
            const int kc = kk + 4 * j + 2 * h;
            v2f t0 = *(const v2f*)(X0 + kc);
            v2f t1 = *(const v2f*)(X1 + kc);
            a0[j] = ok0 ? t0 : (v2f)0.0f;
            a1[j] = ok1 ? t1 : (v2f)0.0f;
        }
#pragma unroll
        for (int j = 0; j < 4; ++j) {
            const int kh = (kk >> 1) + 2 * j + h;   // v2f index = (kk+4j+2h)/2
            // Batch all 8 B fragments (ds_load_b64, all 64 banks hit once),
            // then issue 16 WMMAs: each B fragment feeds both M sub-tiles.
            v2f b[8];
#pragma unroll
            for (int nb = 0; nb < 8; ++nb) {
                b[nb] = ldsW2[(size_t)(nb * 16 + ln) * STRH + kh];
            }
#pragma unroll
            for (int nb = 0; nb < 8; ++nb) {
                acc[nb] = wmma_f32(a0[j], b[nb], acc[nb]);
            }
#pragma unroll
            for (int nb = 0; nb < 8; ++nb) {
                acc[8 + nb] = wmma_f32(a1[j], b[nb], acc[8 + nb]);
            }
        }
    }

    // Store: C/D layout lane ln -> N, VGPR v + 8h -> M.
#pragma unroll
    for (int s = 0; s < 2; ++s) {
#pragma unroll
        for (int nb = 0; nb < 8; ++nb) {
            const int col = nCol0 + nb * 16 + ln;
#pragma unroll
            for (int v = 0; v < 8; ++v) {
                const int row = mBase + s * 16 + v + 8 * h;
                if (row < M) {
                    Y[(size_t)row * DDIM + col] = acc[s * 8 + nb][v];
                }
            }
        }
    }
}

// out[dst[e], :] += Y[src[e], :]   (atomic: multiple edges may share dst)
// 8 edges per block; lane c handles column c. Edge indices are block-uniform
// scalar loads; row read + atomic row update are fully coalesced (1 KB each).
#define EPB 8
extern "C" __global__ __launch_bounds__(256)
void rgcn_scatter_add(const float* __restrict__ Y,
                      const int*   __restrict__ edges,  // [E][2] (src,dst)
                      float* __restrict__ out,
                      int E)
{
    const int e0 = blockIdx.x * EPB;
    const int c  = threadIdx.x;
#pragma unroll
    for (int i = 0; i < EPB; ++i) {
        const int e = e0 + i;
        if (e >= E) break;
        const int src = edges[2 * e + 0];
        const int dst = edges[2 * e + 1];
        const float v = Y[(size_t)src * DDIM + c];
        unsafeAtomicAdd(out + (size_t)dst * DDIM + c, v);  // global_atomic_add_f32
    }
}

extern "C" void kernel_launch(void* const* d_in, const int* in_sizes, int n_in,
                              void* d_out, int out_size, void* d_ws, size_t ws_size,
                              hipStream_t stream)
{
    // setup_inputs order: x, W_rel, W_self, edges
    const float* x      = (const float*)d_in[0];   // [1, 50000, 256] f32
    const float* W_rel  = (const float*)d_in[1];   // [8, 256, 256]   f32
    const float* W_self = (const float*)d_in[2];   // [256, 256]      f32
    const int*   edges  = (const int*)  d_in[3];   // [8, 200000, 2]  i32

    float* out = (float*)d_out;                    // [50000, 256] f32
    float* y   = (float*)d_ws;                     // 51.2 MB scratch

    const int M = in_sizes[0] / DDIM;              // 50000 nodes
    const int R = in_sizes[1] / (DDIM * DDIM);     // 8 relations
    const int E = in_sizes[3] / (2 * R);           // 200000 edges per relation

    const dim3 gemmGrid((M + 255) / 256, 2);       // x: M tiles, y: N half
    const int  scatterGrid = (E + EPB - 1) / EPB;

    // out = x @ W_self  (plain stores: fully initializes d_out)
    rgcn_gemm_wmma<<<gemmGrid, 256, LDS_BYTES, stream>>>(x, W_self, out, M);

    // Per relation: y = x @ W_rel[r]; out[dst] += y[src]
    for (int r = 0; r < R; ++r) {
        rgcn_gemm_wmma<<<gemmGrid, 256, LDS_BYTES, stream>>>(
            x, W_rel + (size_t)r * DDIM * DDIM, y, M);
        rgcn_scatter_add<<<scatterGrid, 256, 0, stream>>>(
            y, edges + (size_t)r * E * 2, out, E);
    }
}